// FocusPooling2D_87840671138089
// MI455X (gfx1250) — compile-verified
//
#include <hip/hip_runtime.h>
#include <cstdint>

// FocusPooling2D for fixed shapes: in [16,256,256,64] f32 -> out [16,205,205,64] f32
// KH=KW=2, keep = round(256*0.3) = 77, nblk = 128, Wout = 77*2 + 51 = 205.

#define B_   16
#define H_   256
#define W_   256
#define C_   64
#define NBLK 128
#define KEEP 77
#define WOUT 205

// ---------------------------------------------------------------------------
// K1: per-column std over H (population), pairwise block max -> scores1[B][C][128]
// Streams the input through LDS with CDNA5 async global->LDS loads,
// double-buffered (32 rows x 1KB per chunk, 64KB LDS total).
// One workgroup per (b, strip of 4 w-columns): strip row = 4*64 floats = 1KB.
// ---------------------------------------------------------------------------
__global__ __launch_bounds__(256) void k1_stats(const float* __restrict__ in,
                                                float* __restrict__ scores1) {
  __shared__ float tile[2][32 * 256];   // 2 x 32KB
  const int b     = blockIdx.y;
  const int strip = blockIdx.x;         // w0 = strip*4
  const int t     = threadIdx.x;        // 256 threads
  const int r     = t >> 3;             // 0..31: row within chunk this thread loads
  const int seg   = t & 7;              // 0..7 : 16B segment group

  const char* gbase =
      (const char*)(in + (size_t)b * H_ * W_ * C_) + (size_t)strip * (4 * C_ * 4);
  // LDS byte offset of tile base (flat shared addr low 32 bits == group offset)
  const unsigned ldsbase = (unsigned)(uintptr_t)(&tile[0][0]);

  auto issue_chunk = [&](int ch, int buf) {
    const char* grow = gbase + (size_t)(ch * 32 + r) * (size_t)(W_ * C_ * 4);
    const unsigned lrow = ldsbase + (unsigned)buf * 32768u + (unsigned)r * 1024u;
#pragma unroll
    for (int j = 0; j < 8; ++j) {
      const unsigned loff = lrow + (unsigned)((seg * 8 + j) * 16);
      const unsigned long long ga =
          (unsigned long long)(uintptr_t)(grow + (seg * 8 + j) * 16);
      // CDNA5 async tensor-path copy: 16B per lane, global -> LDS, no VGPR data
      asm volatile("global_load_async_to_lds_b128 %0, %1, off"
                   :: "v"(loff), "v"(ga) : "memory");
    }
  };

  float s = 0.f, q = 0.f;
  issue_chunk(0, 0);
  for (int ch = 0; ch < 8; ++ch) {
    const int cur = ch & 1;
    if (ch + 1 < 8) {
      issue_chunk(ch + 1, cur ^ 1);                       // prefetch next chunk
      asm volatile("s_wait_asynccnt 8" ::: "memory");     // chunk ch landed (8 outstanding = next chunk)
    } else {
      asm volatile("s_wait_asynccnt 0" ::: "memory");     // drain last chunk
    }
    __syncthreads();
    const float* buf = &tile[cur][0];
#pragma unroll
    for (int rr = 0; rr < 32; ++rr) {
      const float x = buf[rr * 256 + t];  // thread t owns column (wl=t>>6, c=t&63)
      s += x;
      q = fmaf(x, x, q);
    }
    __syncthreads();  // protect buffer before it is overwritten two chunks later
  }
  const float mean = s * (1.f / 256.f);
  const float var  = fmaxf(q * (1.f / 256.f) - mean * mean, 0.f);
  const float sd   = sqrtf(var);

  float* red = &tile[0][0];
  red[t] = sd;
  __syncthreads();
  const int wl = t >> 6;                 // local w 0..3
  if ((wl & 1) == 0) {                   // one thread per block-of-2 columns
    const int c   = t & 63;
    const int blk = strip * 2 + (wl >> 1);
    scores1[((size_t)b * C_ + c) * NBLK + blk] = fmaxf(red[t], red[t + 64]);
  }
}

// ---------------------------------------------------------------------------
// K2: per (b,c): stable descending rank; cond[rank(j)]=1 for j<KEEP mirrors the
// reference's "perm[i] < keep" mask; exclusive prefix-sum of widths.
// table[b][blk][c] = (off<<1)|cond
// ---------------------------------------------------------------------------
__global__ __launch_bounds__(128) void k2_rank(const float* __restrict__ scores,
                                               int* __restrict__ table) {
  const int bc = blockIdx.x;     // b*64 + c
  const int j  = threadIdx.x;    // block id 0..127
  __shared__ float s[NBLK];
  __shared__ int   cond[NBLK];
  __shared__ int   off[NBLK];
  s[j]    = scores[(size_t)bc * NBLK + j];
  cond[j] = 0;
  __syncthreads();
  const float sj = s[j];
  int r = 0;
  for (int i = 0; i < NBLK; ++i) {
    const float si = s[i];
    r += (int)((si > sj) || (si == sj && i < j));  // stable descending rank
  }
  if (j < KEEP) cond[r] = 1;     // ranks are a bijection -> no races
  __syncthreads();
  if (j == 0) {
    int run = 0;
    for (int i = 0; i < NBLK; ++i) { off[i] = run; run += cond[i] ? 2 : 1; }
  }
  __syncthreads();
  const int b = bc >> 6, c = bc & 63;
  table[((size_t)b * NBLK + j) * C_ + c] = (off[j] << 1) | cond[j];
}

// ---------------------------------------------------------------------------
// K3: invert the forward table into a gather map:
// inv[b][pos][c] = (src<<1)|pooled  (pooled: max over src, src+1)
// Block widths partition [0,WOUT), so every slot is written exactly once.
// ---------------------------------------------------------------------------
__global__ __launch_bounds__(256) void k3_inv(const int* __restrict__ table,
                                              int* __restrict__ inv) {
  const int t = blockIdx.x * blockDim.x + threadIdx.x;
  if (t >= B_ * NBLK * C_) return;
  const int c = t & 63, blk = (t >> 6) & 127, b = t >> 13;
  const int p = table[t];
  const int o = p >> 1, cond = p & 1;
  int* dst = inv + ((size_t)b * WOUT + o) * C_ + c;
  if (cond) { dst[0] = (2 * blk) << 1; dst[C_] = (2 * blk + 1) << 1; }
  else      { dst[0] = ((2 * blk) << 1) | 1; }
}

// ---------------------------------------------------------------------------
// K4: stage-2 scores. std over the (permuted) 205-wide stage-1 row is
// permutation-invariant, so iterate blocks in natural order using only cond1:
// raw block -> both columns contribute; pooled block -> max contributes once.
// Fully coalesced second pass over the input. scores2[B][C][128] over h-pairs.
// ---------------------------------------------------------------------------
__global__ __launch_bounds__(256) void k4_stats2(const float* __restrict__ in,
                                                 const int* __restrict__ table1,
                                                 float* __restrict__ scores2) {
  const int t = blockIdx.x * blockDim.x + threadIdx.x;  // B*128*64
  if (t >= B_ * NBLK * C_) return;
  const int c = t & 63, hb = (t >> 6) & 127, b = t >> 13;
  const float* row0 = in + ((size_t)(b * H_ + 2 * hb) * W_) * C_ + c;
  const float* row1 = row0 + (size_t)W_ * C_;
  const int*   tb   = table1 + (size_t)b * NBLK * C_ + c;
  float sa = 0.f, qa = 0.f, sb = 0.f, qb = 0.f;
  for (int blk = 0; blk < NBLK; ++blk) {
    if (blk + 8 < NBLK) {  // gfx1250 global_prefetch_b8
      __builtin_prefetch(row0 + (size_t)(2 * (blk + 8)) * C_, 0, 0);
      __builtin_prefetch(row1 + (size_t)(2 * (blk + 8)) * C_, 0, 0);
    }
    const int   cond = tb[(size_t)blk * C_] & 1;
    const float x0 = row0[(size_t)(2 * blk) * C_];
    const float x1 = row0[(size_t)(2 * blk + 1) * C_];
    const float y0 = row1[(size_t)(2 * blk) * C_];
    const float y1 = row1[(size_t)(2 * blk + 1) * C_];
    if (cond) {
      sa += x0 + x1; qa = fmaf(x0, x0, fmaf(x1, x1, qa));
      sb += y0 + y1; qb = fmaf(y0, y0, fmaf(y1, y1, qb));
    } else {
      const float mx = fmaxf(x0, x1), my = fmaxf(y0, y1);
      sa += mx; qa = fmaf(mx, mx, qa);
      sb += my; qb = fmaf(my, my, qb);
    }
  }
  const float invn = 1.f / (float)WOUT;
  const float ma = sa * invn, mb = sb * invn;
  const float sda = sqrtf(fmaxf(qa * invn - ma * ma, 0.f));
  const float sdb = sqrtf(fmaxf(qb * invn - mb * mb, 0.f));
  scores2[((size_t)b * C_ + c) * NBLK + hb] = fmaxf(sda, sdb);
}

// ---------------------------------------------------------------------------
// K7: final pure-gather output. out[b][hout][w1][c] via inv2 (rows) x inv1 (cols).
// Dense coalesced writes; per-channel-scattered reads stay in the 16MiB/b L2 slice.
// ---------------------------------------------------------------------------
__global__ __launch_bounds__(256) void k7_out(const float* __restrict__ in,
                                              const int* __restrict__ inv1,
                                              const int* __restrict__ inv2,
                                              float* __restrict__ out) {
  const long long TOTAL = (long long)B_ * WOUT * WOUT * C_;
  long long t = (long long)blockIdx.x * blockDim.x + threadIdx.x;
  if (t >= TOTAL) return;
  const int c = (int)(t & 63);
  long long rest = t >> 6;
  const int w1   = (int)(rest % WOUT); rest /= WOUT;
  const int hout = (int)(rest % WOUT);
  const int b    = (int)(rest / WOUT);
  const int e2 = inv2[((size_t)b * WOUT + hout) * C_ + c];
  const int e1 = inv1[((size_t)b * WOUT + w1) * C_ + c];
  const int h0 = e2 >> 1, p2 = e2 & 1;
  const int w0 = e1 >> 1, p1 = e1 & 1;
  const float* base = in + ((size_t)(b * H_ + h0) * W_ + w0) * C_ + c;
  float va = base[0];
  if (p1) va = fmaxf(va, base[C_]);
  float v = va;
  if (p2) {
    const float* base2 = base + (size_t)W_ * C_;
    float vb = base2[0];
    if (p1) vb = fmaxf(vb, base2[C_]);
    v = fmaxf(v, vb);
  }
  out[t] = v;
}

// ---------------------------------------------------------------------------
extern "C" void kernel_launch(void* const* d_in, const int* in_sizes, int n_in,
                              void* d_out, int out_size, void* d_ws, size_t ws_size,
                              hipStream_t stream) {
  const float* in  = (const float*)d_in[0];
  float*       out = (float*)d_out;
  char*        ws  = (char*)d_ws;

  // workspace layout (~3.6 MB total)
  float* scores1 = (float*)(ws + 0);        // 16*64*128*4 = 524288
  int*   table1  = (int*)  (ws + 524288);   // 524288
  int*   inv1    = (int*)  (ws + 1048576);  // 16*205*64*4 = 839680
  float* scores2 = (float*)(ws + 1888256);  // 524288
  int*   table2  = (int*)  (ws + 2412544);  // 524288
  int*   inv2    = (int*)  (ws + 2936832);  // 839680

  k1_stats<<<dim3(W_ / 4, B_), 256, 0, stream>>>(in, scores1);
  k2_rank <<<B_ * C_, 128, 0, stream>>>(scores1, table1);
  k3_inv  <<<(B_ * NBLK * C_) / 256, 256, 0, stream>>>(table1, inv1);
  k4_stats2<<<(B_ * NBLK * C_) / 256, 256, 0, stream>>>(in, table1, scores2);
  k2_rank <<<B_ * C_, 128, 0, stream>>>(scores2, table2);
  k3_inv  <<<(B_ * NBLK * C_) / 256, 256, 0, stream>>>(table2, inv2);

  const long long total = (long long)B_ * WOUT * WOUT * C_;
  k7_out<<<(unsigned)((total + 255) / 256), 256, 0, stream>>>(in, inv1, inv2, out);
}